// MMSolver_87746181857521
// MI455X (gfx1250) — compile-verified
//
#include <hip/hip_runtime.h>

// MI455X (gfx1250) LLG micromagnetics solver.
// One kernel launch per full RK4 step (halo-4 temporal blocking in LDS),
// exchange Laplacian computed on the matrix pipe via V_WMMA_F32_16X16X4_F32.
// Stage-invariant fields (B_ext + s*src, 2A/(Ms*dx^2), mu0*Ms) are precomputed
// once per step into LDS, so the four RK4 stages are pure LDS+VALU+WMMA.

typedef float v2f __attribute__((ext_vector_type(2)));
typedef float v8f __attribute__((ext_vector_type(8)));

#define NXY    256
#define NCELL  65536
#define TILE   40
#define PITCH  41
#define TPP    (TILE * PITCH)      // 1640 floats per component plane
#define NT     256                 // threads per workgroup (8 waves)
#define NWAVE  8
#define RELAX_STEPS 100

#define GAMMA_LL 1.7595e11f
#define DT_S     5e-12f
#define HSTEP    (GAMMA_LL * DT_S)     // 0.87975
#define AEXCH    3.65e-12f
#define MU0F     1.2566371e-6f         // float32(4e-7*pi)
#define INVDX2   4.0e14f               // 1/(50nm)^2

__device__ __forceinline__ int clampi(int v, int lo, int hi) {
  return v < lo ? lo : (v > hi ? hi : v);
}

// Banded 1D Laplacian matrix entry: band(d)=1 at d==0,2 ; -2 at d==1 ; else 0.
__device__ __forceinline__ float bandf(int d) {
  return (d == 0 || d == 2) ? 1.0f : ((d == 1) ? -2.0f : 0.0f);
}

// Compute LLG torque over region [s, 40-s)^2 of the tile.
// Laplacian of each m-component done as Lr*M + M*Lc with chained
// v_wmma_f32_16x16x4_f32 (K padded 18->20, clamped reads carry zero coeff).
// Bf: 5 precomputed LDS planes {Btx, Bty, Btz, ec, dg}.
__device__ __forceinline__ void stage_torque(
    const float* __restrict__ IN,   // LDS stage input  [3][TPP]
    float*       __restrict__ TQ,   // LDS torque out   [3][TPP]
    const float* __restrict__ Bf,   // LDS fields       [5][TPP]
    int s, int S, int nb,
    int wave, int hi, int n,
    float alpha, float inva)
{
  for (int b = wave; b < nb * nb; b += NWAVE) {   // wave-uniform loop: EXEC all 1s
    const int ir = b / nb, ic = b % nb;
    const int r0 = (ir == 0) ? s : ((ir == 1) ? s + 16 : s + S - 16);
    const int c0 = (ic == 0) ? s : ((ic == 1) ? s + 16 : s + S - 16);

    float lap[3][8];
#pragma unroll
    for (int comp = 0; comp < 3; ++comp) {
      const float* M = IN + comp * TPP;
      v8f acc = {0.f, 0.f, 0.f, 0.f, 0.f, 0.f, 0.f, 0.f};
      // row part: Lr (16x20 const, A operand) * M-rows (20x16, B operand)
#pragma unroll
      for (int kc = 0; kc < 5; ++kc) {
        const int kk = 4 * kc + 2 * hi;            // K for vgpr0; vgpr1 = kk+1
        v2f a, bb;
        a.x = bandf(kk - n);                       // Lr[mo=n][kk]
        a.y = bandf(kk + 1 - n);
        int rA = r0 - 1 + kk;     if (rA > TILE - 1) rA = TILE - 1;  // pad: coeff 0
        int rB = rA + 1;          if (rB > TILE - 1) rB = TILE - 1;
        bb.x = M[rA * PITCH + c0 + n];             // B[K=kk][N=n]
        bb.y = M[rB * PITCH + c0 + n];
        acc = __builtin_amdgcn_wmma_f32_16x16x4_f32(false, a, false, bb,
                                                    (short)0, acc, false, false);
      }
      // col part: M-cols (16x20, A operand) * Lc (20x16 const, B operand)
#pragma unroll
      for (int kc = 0; kc < 5; ++kc) {
        const int kk = 4 * kc + 2 * hi;
        v2f a, bb;
        int cA = c0 - 1 + kk;     if (cA > TILE - 1) cA = TILE - 1;
        int cB = cA + 1;          if (cB > TILE - 1) cB = TILE - 1;
        const int rowM = (r0 + n) * PITCH;         // A[M=n][K=kk]
        a.x = M[rowM + cA];
        a.y = M[rowM + cB];
        bb.x = bandf(kk - n);                      // Lc[kk][no=n]
        bb.y = bandf(kk + 1 - n);
        acc = __builtin_amdgcn_wmma_f32_16x16x4_f32(false, a, false, bb,
                                                    (short)0, acc, false, false);
      }
#pragma unroll
      for (int v = 0; v < 8; ++v) lap[comp][v] = acc[v];
    }

    // Per-cell LLG torque (VALU, co-executes with matrix pipe of other waves).
#pragma unroll
    for (int v = 0; v < 8; ++v) {
      const int ra = r0 + v + 8 * hi;              // D layout: M = v + 8*(lane>=16)
      const int ca = c0 + n;                       //           N = lane & 15
      const int o  = ra * PITCH + ca;
      const float mx = IN[o], my = IN[TPP + o], mz = IN[2 * TPP + o];
      const float ec = Bf[3 * TPP + o];            // 2A/(Ms*dx^2)
      const float Bx = Bf[o]           + ec * lap[0][v];
      const float By = Bf[TPP + o]     + ec * lap[1][v];
      const float Bz = Bf[2 * TPP + o] + ec * lap[2][v] - Bf[4 * TPP + o] * mz;
      const float cx = my * Bz - mz * By;          // m x B
      const float cy = mz * Bx - mx * Bz;
      const float cz = mx * By - my * Bx;
      const float qx = my * cz - mz * cy;          // m x (m x B)
      const float qy = mz * cx - mx * cz;
      const float qz = mx * cy - my * cx;
      // overlapping blocks write bit-identical values -> benign
      TQ[o]           = -(cx + alpha * qx) * inva;
      TQ[TPP + o]     = -(cy + alpha * qy) * inva;
      TQ[2 * TPP + o] = -(cz + alpha * qz) * inva;
    }
  }
}

__global__ __launch_bounds__(NT)
void mm_step(const float* __restrict__ m_in, float* __restrict__ m_out,
             const float* __restrict__ Bext, const float* __restrict__ srcm,
             const float* __restrict__ Msat, const float* __restrict__ sig,
             const float* __restrict__ m0,   const float* __restrict__ pmask,
             float* __restrict__ part, int t, float alpha)
{
  __shared__ float sm[3 * TPP];    // original m (persists all stages)
  __shared__ float bufX[3 * TPP];  // stage torque
  __shared__ float bufY[3 * TPP];  // stage input mt
  __shared__ float Bf[5 * TPP];    // Btx, Bty, Btz, ec, dg (stage-invariant)
  __shared__ float red[NT];

  const int tid  = threadIdx.x;
  const int wave = tid >> 5, lane = tid & 31;
  const int hi   = (lane >> 4) & 1, n = lane & 15;
  const int gx0  = (int)blockIdx.x * 32 - 4;
  const int gy0  = (int)blockIdx.y * 32 - 4;

  float ssig = 0.0f;
  if (t >= 0) ssig = sig[t];
  const float inva = 1.0f / (1.0f + alpha * alpha);

  // Load m tile with clamped (Neumann replicate) halo; precompute field planes.
  for (int idx = tid; idx < TILE * TILE; idx += NT) {
    const int r = idx / TILE, c = idx - r * TILE;
    const int gx = clampi(gx0 + r, 0, NXY - 1);
    const int gy = clampi(gy0 + c, 0, NXY - 1);
    const int g  = gx * NXY + gy;
    const int o  = r * PITCH + c;
    sm[o]           = m_in[g];
    sm[TPP + o]     = m_in[NCELL + g];
    sm[2 * TPP + o] = m_in[2 * NCELL + g];
    const float Ms  = Msat[g];
    Bf[o]           = Bext[g]             + ssig * srcm[g];
    Bf[TPP + o]     = Bext[NCELL + g]     + ssig * srcm[NCELL + g];
    Bf[2 * TPP + o] = Bext[2 * NCELL + g] + ssig * srcm[2 * NCELL + g];
    Bf[3 * TPP + o] = (2.0f * AEXCH * INVDX2) / Ms;
    Bf[4 * TPP + o] = MU0F * Ms;
  }
  for (int idx = tid; idx < 3 * TPP; idx += NT) { bufX[idx] = 0.f; bufY[idx] = 0.f; }
  __syncthreads();

  float kr[12];                     // RK4 accumulator: 4 interior cells x 3 comps
#pragma unroll
  for (int i = 0; i < 12; ++i) kr[i] = 0.f;

  const float h = HSTEP;
  const float kwt[4] = {1.f, 2.f, 2.f, 1.f};
  const float mtc[4] = {0.5f * h, 0.5f * h, h, 0.f};

#pragma unroll
  for (int st = 0; st < 4; ++st) {
    const int s  = st + 1;
    const int S  = TILE - 2 * s;            // 38,36,34,32 (compile-time)
    const int nb = (S > 32) ? 3 : 2;
    const float* INp = (st == 0) ? sm : bufY;
    stage_torque(INp, bufX, Bf, s, S, nb, wave, hi, n, alpha, inva);
    __syncthreads();
    // Fused epilogue: RK4 accumulate (reads bufX interior) + mt transform
    // (reads bufX region, writes bufY region) -- disjoint write target.
#pragma unroll
    for (int j = 0; j < 4; ++j) {
      const int id = tid + j * NT;          // 0..1023 over 32x32 interior
      const int o  = (4 + (id >> 5)) * PITCH + (4 + (id & 31));
      kr[j]     += kwt[st] * bufX[o];
      kr[4 + j] += kwt[st] * bufX[TPP + o];
      kr[8 + j] += kwt[st] * bufX[2 * TPP + o];
    }
    if (st < 3) {
      for (int idx = tid; idx < S * S; idx += NT) {
        const int r = s + idx / S, c = s + idx % S;   // S const: folds to mul-shift
        const int o = r * PITCH + c;
        bufY[o]           = sm[o]           + mtc[st] * bufX[o];
        bufY[TPP + o]     = sm[TPP + o]     + mtc[st] * bufX[TPP + o];
        bufY[2 * TPP + o] = sm[2 * TPP + o] + mtc[st] * bufX[2 * TPP + o];
      }
      __syncthreads();
    }
  }

  // Final update + probe partial over this WG's 32x32 interior.
  float partial = 0.f;
  const float s6 = h / 6.0f;
#pragma unroll
  for (int j = 0; j < 4; ++j) {
    const int id = tid + j * NT;
    const int r = 4 + (id >> 5), c = 4 + (id & 31);
    const int o = r * PITCH + c;
    const int g = (gx0 + r) * NXY + (gy0 + c);   // interior -> always in range
    const float mx = sm[o]           + s6 * kr[j];
    const float my = sm[TPP + o]     + s6 * kr[4 + j];
    const float mz = sm[2 * TPP + o] + s6 * kr[8 + j];
    m_out[g]             = mx;
    m_out[NCELL + g]     = my;
    m_out[2 * NCELL + g] = mz;
    if (t >= 0) partial += (mx - m0[g]) * Msat[g] * pmask[g];
  }
  if (t >= 0) {
    red[tid] = partial;
    __syncthreads();
    for (int off = NT / 2; off > 0; off >>= 1) {
      if (tid < off) red[tid] += red[tid + off];
      __syncthreads();
    }
    if (tid == 0) part[t * 64 + ((int)blockIdx.y * 8 + (int)blockIdx.x)] = red[0];
  }
}

__global__ void mm_init(float* __restrict__ mA) {
  const int idx = blockIdx.x * blockDim.x + threadIdx.x;
  if (idx < NCELL) {
    mA[idx] = 0.f; mA[NCELL + idx] = 0.f; mA[2 * NCELL + idx] = 1.f;
  }
}

__global__ void mm_pmsum(const float* __restrict__ pmask, float* __restrict__ pmsum) {
  __shared__ float red[256];
  const int tid = threadIdx.x;
  float s = 0.f;
  for (int i = tid; i < NCELL; i += 256) s += pmask[i];
  red[tid] = s;
  __syncthreads();
  for (int off = 128; off > 0; off >>= 1) {
    if (tid < off) red[tid] += red[tid + off];
    __syncthreads();
  }
  if (tid == 0) pmsum[0] = red[0];
}

__global__ void mm_scale(const float* __restrict__ part, const float* __restrict__ pmsum,
                         float* __restrict__ out, int T) {
  const int t = blockIdx.x * blockDim.x + threadIdx.x;
  if (t < T) {
    float s = 0.f;
    for (int w = 0; w < 64; ++w) s += part[t * 64 + w];   // fixed order: deterministic
    out[t] = s / pmsum[0];
  }
}

extern "C" void kernel_launch(void* const* d_in, const int* in_sizes, int n_in,
                              void* d_out, int out_size, void* d_ws, size_t ws_size,
                              hipStream_t stream) {
  (void)n_in; (void)out_size; (void)ws_size;
  const float* sig   = (const float*)d_in[0];   // (1,T,1)
  const float* Bext  = (const float*)d_in[1];   // (1,3,256,256)
  const float* Msat  = (const float*)d_in[2];   // (1,1,256,256)
  const float* srcm  = (const float*)d_in[3];   // (1,3,256,256)
  const float* pmask = (const float*)d_in[4];   // (256,256)
  const int T = in_sizes[0];

  float* ws    = (float*)d_ws;
  float* mA    = ws;                 // 3*NCELL
  float* mB    = ws + 3 * NCELL;     // 3*NCELL
  float* m0    = ws + 6 * NCELL;     // 3*NCELL
  float* part  = ws + 9 * NCELL;     // T*64 (<= NCELL)
  float* pmsum = ws + 10 * NCELL;    // 1
  float* out   = (float*)d_out;

  mm_init <<<dim3(256), dim3(256), 0, stream>>>(mA);
  mm_pmsum<<<dim3(1),   dim3(256), 0, stream>>>(pmask, pmsum);

  const dim3 grid(8, 8), blk(NT);
  float* cur = mA; float* nxt = mB;
  for (int i = 0; i < RELAX_STEPS; ++i) {
    mm_step<<<grid, blk, 0, stream>>>(cur, nxt, Bext, srcm, Msat, sig,
                                      m0, pmask, part, -1, 0.5f);
    float* tmp = cur; cur = nxt; nxt = tmp;
  }
  hipMemcpyAsync(m0, cur, 3 * NCELL * sizeof(float), hipMemcpyDeviceToDevice, stream);
  for (int t = 0; t < T; ++t) {
    mm_step<<<grid, blk, 0, stream>>>(cur, nxt, Bext, srcm, Msat, sig,
                                      m0, pmask, part, t, 0.01f);
    float* tmp = cur; cur = nxt; nxt = tmp;
  }
  mm_scale<<<dim3((T + 255) / 256), dim3(256), 0, stream>>>(part, pmsum, out, T);
}